// DAGCNBert_82678120448020
// MI455X (gfx1250) — compile-verified
//
#include <hip/hip_runtime.h>
#include <hip/hip_bf16.h>

// ---------- Problem constants (from reference) ----------
#define B_  16
#define S_  512
#define D_  768
#define H_  12
#define DK_ 64
#define MROWS (B_ * S_)   // 8192

// ---------- CDNA5 WMMA f32 16x16x4 ----------
typedef __attribute__((ext_vector_type(2))) float v2f;
typedef __attribute__((ext_vector_type(8))) float v8f;

__device__ __forceinline__ v8f wmma_f32x4(v2f a, v2f b, v8f c) {
    return __builtin_amdgcn_wmma_f32_16x16x4_f32(
        false, a, false, b, (short)0, c, false, false);
}

__device__ __forceinline__ v8f v8f_zero() {
    v8f z = {0.f, 0.f, 0.f, 0.f, 0.f, 0.f, 0.f, 0.f};
    return z;
}

// ---------- CDNA5 async global->LDS (ASYNCcnt path, inline asm) ----------
// VDST = LDS byte address VGPR (low 32 bits of a generic shared pointer are
// the LDS offset per the flat-addressing rules), VADDR = 64-bit global addr.
__device__ __forceinline__ void async_copy16(const float* g, float* l) {
    asm volatile("global_load_async_to_lds_b128 %0, %1, off"
                 :: "v"((unsigned)(unsigned long long)l), "v"(g)
                 : "memory");
}

#if defined(__has_builtin)
#if __has_builtin(__builtin_amdgcn_s_wait_asynccnt)
#define HAVE_WAIT_ASYNC 1
#endif
#endif

__device__ __forceinline__ void wait_async0() {
#ifdef HAVE_WAIT_ASYNC
    __builtin_amdgcn_s_wait_asynccnt(0);
#else
    asm volatile("s_wait_asynccnt 0x0" ::: "memory");
#endif
}

// =====================================================================
// Kernel 0: 32x32 tiled transpose, Wt[N][K] = W[K][N]
// =====================================================================
__global__ __launch_bounds__(256)
void transpose_kernel(const float* __restrict__ W, float* __restrict__ Wt,
                      int K, int N) {
    __shared__ float t[32][33];
    const int k0 = blockIdx.x * 32, n0 = blockIdx.y * 32;
    const int tx = threadIdx.x & 31, ty = threadIdx.x >> 5;
    for (int r = ty; r < 32; r += 8)
        t[r][tx] = W[(size_t)(k0 + r) * N + (n0 + tx)];
    __syncthreads();
    for (int r = ty; r < 32; r += 8)
        Wt[(size_t)(n0 + r) * K + (k0 + tx)] = t[tx][r];
}

// =====================================================================
// WMMA GEMM: C[M,N] = A[M,K] @ Wt[N,K]^T + bias  (Wt pre-transposed)
// 64x64 block tile, 4 waves, async global->LDS staging.
// LDS row stride 20 floats -> bank-conflict-free b64 fragment loads.
// =====================================================================
#define TM 64
#define TN 64
#define TK 16
#define LSTRIDE 20

__global__ __launch_bounds__(128)
void gemm_bias_kernel(const float* __restrict__ A, const float* __restrict__ Wt,
                      const float* __restrict__ bias, float* __restrict__ C,
                      int M, int N, int K) {
    __shared__ float As[TM * LSTRIDE];
    __shared__ float Bs[TN * LSTRIDE];
    const int tid  = threadIdx.x;
    const int wid  = tid >> 5;
    const int lane = tid & 31;
    const int half = lane >> 4;
    const int l16  = lane & 15;
    const int m0 = blockIdx.x * TM;
    const int n0 = blockIdx.y * TN;

    v8f acc[4];
    for (int i = 0; i < 4; ++i) acc[i] = v8f_zero();

    for (int k0 = 0; k0 < K; k0 += TK) {
        // 64 rows x 16 floats per tile = 256 b128 chunks; 2 per thread each side
        for (int i = tid; i < 256; i += 128) {
            const int r = i >> 2, ch = (i & 3) * 4;
            async_copy16(A  + (size_t)(m0 + r) * K + k0 + ch, &As[r * LSTRIDE + ch]);
            async_copy16(Wt + (size_t)(n0 + r) * K + k0 + ch, &Bs[r * LSTRIDE + ch]);
        }
        if (k0 + TK < K)  // prefetch next A tile rows toward L2
            __builtin_prefetch(A + (size_t)(m0 + (tid >> 1)) * K + k0 + TK, 0, 1);
        wait_async0();
        __syncthreads();

        const float* arow = &As[(wid * 16 + l16) * LSTRIDE];
        for (int kk = 0; kk < TK; kk += 4) {
            v2f a = *(const v2f*)&arow[kk + 2 * half];
            for (int nt = 0; nt < 4; ++nt) {
                v2f bf = *(const v2f*)&Bs[(nt * 16 + l16) * LSTRIDE + kk + 2 * half];
                acc[nt] = wmma_f32x4(a, bf, acc[nt]);
            }
        }
        __syncthreads();
    }

    const int mbase = m0 + wid * 16;
    for (int nt = 0; nt < 4; ++nt) {
        int n = n0 + nt * 16 + l16;
        float bv = bias[n];
        for (int r = 0; r < 8; ++r)
            C[(size_t)(mbase + r + 8 * half) * N + n] = acc[nt][r] + bv;
    }
}

// =====================================================================
// Fused output: C = relu(A1@W1t^T + A2@W2t^T + bias)
// =====================================================================
__global__ __launch_bounds__(128)
void out_gemm_kernel(const float* __restrict__ A1, const float* __restrict__ W1t,
                     const float* __restrict__ A2, const float* __restrict__ W2t,
                     const float* __restrict__ bias, float* __restrict__ C,
                     int M, int N, int K) {
    __shared__ float As[TM * LSTRIDE];
    __shared__ float Bs[TN * LSTRIDE];
    const int tid  = threadIdx.x;
    const int wid  = tid >> 5;
    const int lane = tid & 31;
    const int half = lane >> 4;
    const int l16  = lane & 15;
    const int m0 = blockIdx.x * TM;
    const int n0 = blockIdx.y * TN;

    v8f acc[4];
    for (int i = 0; i < 4; ++i) acc[i] = v8f_zero();

    for (int pass = 0; pass < 2; ++pass) {
        const float* A = pass ? A2 : A1;
        const float* W = pass ? W2t : W1t;
        for (int k0 = 0; k0 < K; k0 += TK) {
            for (int i = tid; i < 256; i += 128) {
                const int r = i >> 2, ch = (i & 3) * 4;
                async_copy16(A + (size_t)(m0 + r) * K + k0 + ch, &As[r * LSTRIDE + ch]);
                async_copy16(W + (size_t)(n0 + r) * K + k0 + ch, &Bs[r * LSTRIDE + ch]);
            }
            wait_async0();
            __syncthreads();
            const float* arow = &As[(wid * 16 + l16) * LSTRIDE];
            for (int kk = 0; kk < TK; kk += 4) {
                v2f a = *(const v2f*)&arow[kk + 2 * half];
                for (int nt = 0; nt < 4; ++nt) {
                    v2f bf = *(const v2f*)&Bs[(nt * 16 + l16) * LSTRIDE + kk + 2 * half];
                    acc[nt] = wmma_f32x4(a, bf, acc[nt]);
                }
            }
            __syncthreads();
        }
    }

    const int mbase = m0 + wid * 16;
    for (int nt = 0; nt < 4; ++nt) {
        int n = n0 + nt * 16 + l16;
        float bv = bias[n];
        for (int r = 0; r < 8; ++r)
            C[(size_t)(mbase + r + 8 * half) * N + n] = fmaxf(acc[nt][r] + bv, 0.f);
    }
}

// =====================================================================
// aspect: asp = aspect@Wd + bd ; aw = asp@weight_m[h] ;
//         aspect_scores[b,h,t] = tanh(aw . K[b,h,t,:] + bias_m)
// =====================================================================
__global__ __launch_bounds__(128)
void aspect_kernel(const float* __restrict__ aspect, const float* __restrict__ Wd,
                   const float* __restrict__ bd, const float* __restrict__ weight_m,
                   const float* __restrict__ bias_m, const float* __restrict__ Kws,
                   float* __restrict__ asp_sc) {
    const int h = blockIdx.x, b = blockIdx.y;
    const int tid = threadIdx.x;
    __shared__ float asp[DK_];
    __shared__ float aw[DK_];

    if (tid < DK_) {
        float acc = bd[tid];
        const float* av = aspect + (size_t)b * D_;
        for (int d = 0; d < D_; ++d) acc += av[d] * Wd[(size_t)d * DK_ + tid];
        asp[tid] = acc;
    }
    __syncthreads();
    if (tid < DK_) {
        float acc = 0.f;
        const float* wm = weight_m + (size_t)h * DK_ * DK_;
        for (int e = 0; e < DK_; ++e) acc += asp[e] * wm[(size_t)e * DK_ + tid];
        aw[tid] = acc;
    }
    __syncthreads();
    const float bm = bias_m[0];
    for (int t = tid; t < S_; t += 128) {
        const float* kr = Kws + ((size_t)(b * S_ + t)) * D_ + h * DK_;
        float acc = bm;
        for (int e = 0; e < DK_; ++e) acc += aw[e] * kr[e];
        asp_sc[((size_t)b * H_ + h) * S_ + t] = tanhf(acc);
    }
}

// =====================================================================
// Fused scores (WMMA QK^T) + aspect bias + mask + softmax + head mean.
// grid (S/16, B), 4 waves: split t-chunks per head, then softmax rows.
// =====================================================================
__global__ __launch_bounds__(128)
void scores_softmax_avg_kernel(const float* __restrict__ Qws, const float* __restrict__ Kws,
                               const float* __restrict__ asp_sc, const int* __restrict__ mask,
                               float* __restrict__ attn_avg) {
    const int b  = blockIdx.y;
    const int s0 = blockIdx.x * 16;
    const int tid  = threadIdx.x;
    const int wid  = tid >> 5;
    const int lane = tid & 31;
    const int half = lane >> 4;
    const int l16  = lane & 15;

    __shared__ float sc[16][S_];
    __shared__ float accum[16][S_];

    for (int i = tid; i < 16 * S_; i += 128) ((float*)accum)[i] = 0.f;
    __syncthreads();

    for (int h = 0; h < H_; ++h) {
        const float2* qrow =
            (const float2*)(Qws + ((size_t)(b * S_ + s0 + l16)) * D_ + h * DK_);
        v2f afrag[16];
        for (int kk = 0; kk < 16; ++kk) {
            float2 t2 = qrow[2 * kk + half];
            afrag[kk].x = t2.x; afrag[kk].y = t2.y;
        }
        for (int tci = 0; tci < 8; ++tci) {
            const int t0 = (wid * 8 + tci) * 16;
            const float2* krow =
                (const float2*)(Kws + ((size_t)(b * S_ + t0 + l16)) * D_ + h * DK_);
            v8f acc = v8f_zero();
            for (int kk = 0; kk < 16; ++kk) {
                float2 t2 = krow[2 * kk + half];
                v2f bf; bf.x = t2.x; bf.y = t2.y;
                acc = wmma_f32x4(afrag[kk], bf, acc);
            }
            const int t = t0 + l16;
            const float addv = asp_sc[((size_t)b * H_ + h) * S_ + t];
            const bool mok = mask[b * S_ + t] != 0;
            for (int r = 0; r < 8; ++r) {
                float v = acc[r] * 0.125f + addv;      // 1/sqrt(64)
                if (!mok) v = -1e18f;
                sc[r + 8 * half][t] = v;
            }
        }
        __syncthreads();
        for (int rr = 0; rr < 4; ++rr) {
            const int r = wid * 4 + rr;
            float v[16];
            float mx = -3.0e38f;
            for (int j = 0; j < 16; ++j) {
                v[j] = sc[r][lane + 32 * j];
                mx = fmaxf(mx, v[j]);
            }
            for (int m = 16; m >= 1; m >>= 1) mx = fmaxf(mx, __shfl_xor(mx, m));
            float sum = 0.f;
            for (int j = 0; j < 16; ++j) { v[j] = __expf(v[j] - mx); sum += v[j]; }
            for (int m = 16; m >= 1; m >>= 1) sum += __shfl_xor(sum, m);
            const float scale = (1.0f / (float)H_) / sum;
            for (int j = 0; j < 16; ++j) accum[r][lane + 32 * j] += v[j] * scale;
        }
        __syncthreads();
    }

    for (int i = tid; i < 16 * S_; i += 128) {
        int r = i >> 9, c = i & (S_ - 1);
        attn_avg[((size_t)(b * S_ + s0 + r)) * S_ + c] = accum[r][c];
    }
}

// =====================================================================
// Per-row top-k (k<=32): one wave per row, iterative wave argmax.
// =====================================================================
__global__ __launch_bounds__(32)
void topk_kernel(const float* __restrict__ attn_avg, const int* __restrict__ topk_ptr,
                 int* __restrict__ idx_out) {
    const int row  = blockIdx.x;
    const int lane = threadIdx.x;
    const float* av = attn_avg + (size_t)row * S_;
    float v[16];
    for (int j = 0; j < 16; ++j) v[j] = av[lane + 32 * j];
    int kmax = topk_ptr[0]; if (kmax > 32) kmax = 32;

    for (int k = 0; k < kmax; ++k) {
        float bestv = -3.0e38f; int besti = 0x7fffffff;
        for (int j = 0; j < 16; ++j) {
            int idx = lane + 32 * j;
            if (v[j] > bestv || (v[j] == bestv && idx < besti)) { bestv = v[j]; besti = idx; }
        }
        for (int m = 16; m >= 1; m >>= 1) {
            float ov = __shfl_xor(bestv, m);
            int   oi = __shfl_xor(besti, m);
            if (ov > bestv || (ov == bestv && oi < besti)) { bestv = ov; besti = oi; }
        }
        if ((besti & 31) == lane) v[besti >> 5] = -3.0e38f;
        if (lane == 0) idx_out[(size_t)row * 32 + k] = besti;
    }
}

// =====================================================================
// Sparse neighbor mean: aggr[b,s,:] = mean over {topk} U {s}
// =====================================================================
__global__ __launch_bounds__(192)
void aggregate_kernel(const float* __restrict__ hidden, const int* __restrict__ idx,
                      const int* __restrict__ topk_ptr, float* __restrict__ aggr) {
    const int row = blockIdx.x;
    const int b = row >> 9, s = row & (S_ - 1);
    const int tid = threadIdx.x;
    int kmax = topk_ptr[0]; if (kmax > 32) kmax = 32;

    __shared__ int sid[32];
    if (tid < 32) sid[tid] = idx[(size_t)row * 32 + tid];
    __syncthreads();

    const float* hb = hidden + (size_t)b * S_ * D_;
    float4 acc = make_float4(0.f, 0.f, 0.f, 0.f);
    bool self_in = false;
    for (int j = 0; j < kmax; ++j) {
        int t = sid[j];
        if (t == s) self_in = true;
        float4 hv = *(const float4*)(hb + (size_t)t * D_ + tid * 4);
        acc.x += hv.x; acc.y += hv.y; acc.z += hv.z; acc.w += hv.w;
    }
    float deg = (float)kmax;
    if (!self_in) {
        float4 hv = *(const float4*)(hb + (size_t)s * D_ + tid * 4);
        acc.x += hv.x; acc.y += hv.y; acc.z += hv.z; acc.w += hv.w;
        deg += 1.f;
    }
    const float inv = 1.f / deg;
    acc.x *= inv; acc.y *= inv; acc.z *= inv; acc.w *= inv;
    *(float4*)(aggr + (size_t)row * D_ + tid * 4) = acc;
}

// =====================================================================
// Launch
// =====================================================================
extern "C" void kernel_launch(void* const* d_in, const int* in_sizes, int n_in,
                              void* d_out, int out_size, void* d_ws, size_t ws_size,
                              hipStream_t stream) {
    const float* hidden   = (const float*)d_in[0];
    const float* aspect   = (const float*)d_in[1];
    const float* Wq       = (const float*)d_in[2];
    const float* bq       = (const float*)d_in[3];
    const float* Wk       = (const float*)d_in[4];
    const float* bk       = (const float*)d_in[5];
    const float* Wd       = (const float*)d_in[6];
    const float* bd       = (const float*)d_in[7];
    const float* weight_m = (const float*)d_in[8];
    const float* bias_m   = (const float*)d_in[9];
    const float* W_agg    = (const float*)d_in[10];
    const float* b_agg    = (const float*)d_in[11];
    const float* W_self   = (const float*)d_in[12];
    const int*   mask     = (const int*)d_in[13];
    const int*   topk     = (const int*)d_in[14];
    float* out = (float*)d_out;

    // Workspace: Q | K | asp_sc | attn_avg | idx | aggr | 4 transposed weights
    float* Qws  = (float*)d_ws;
    float* Kws  = Qws + (size_t)MROWS * D_;
    float* aspS = Kws + (size_t)MROWS * D_;
    float* attn = aspS + (size_t)B_ * H_ * S_;
    int*   idxb = (int*)(attn + (size_t)B_ * S_ * S_);
    float* aggr = (float*)(idxb + (size_t)MROWS * 32);
    float* Wqt  = aggr + (size_t)MROWS * D_;
    float* Wkt  = Wqt + (size_t)D_ * D_;
    float* Wst  = Wkt + (size_t)D_ * D_;
    float* Wat  = Wst + (size_t)D_ * D_;

    dim3 trGrid(D_ / 32, D_ / 32);
    transpose_kernel<<<trGrid, 256, 0, stream>>>(Wq, Wqt, D_, D_);
    transpose_kernel<<<trGrid, 256, 0, stream>>>(Wk, Wkt, D_, D_);
    transpose_kernel<<<trGrid, 256, 0, stream>>>(W_self, Wst, D_, D_);
    transpose_kernel<<<trGrid, 256, 0, stream>>>(W_agg, Wat, D_, D_);

    dim3 gemmGrid(MROWS / TM, D_ / TN);   // 128 x 12
    gemm_bias_kernel<<<gemmGrid, 128, 0, stream>>>(hidden, Wqt, bq, Qws, MROWS, D_, D_);
    gemm_bias_kernel<<<gemmGrid, 128, 0, stream>>>(hidden, Wkt, bk, Kws, MROWS, D_, D_);

    aspect_kernel<<<dim3(H_, B_), 128, 0, stream>>>(aspect, Wd, bd, weight_m, bias_m,
                                                    Kws, aspS);

    scores_softmax_avg_kernel<<<dim3(S_ / 16, B_), 128, 0, stream>>>(Qws, Kws, aspS,
                                                                     mask, attn);

    topk_kernel<<<MROWS, 32, 0, stream>>>(attn, topk, idxb);

    aggregate_kernel<<<MROWS, 192, 0, stream>>>(hidden, idxb, topk, aggr);

    out_gemm_kernel<<<gemmGrid, 128, 0, stream>>>(hidden, Wst, aggr, Wat,
                                                  b_agg, out, MROWS, D_, D_);
}